// Net_21732534518255
// MI455X (gfx1250) — compile-verified
//
#include <hip/hip_runtime.h>
#include <hip/hip_bf16.h>

typedef __attribute__((ext_vector_type(2))) float v2f;
typedef __attribute__((ext_vector_type(8))) float v8f;

// ---------------------------------------------------------------------------
// Conv stack: one wave (32 lanes) per (sample, group). Entire conv pipeline
// lives in per-wave LDS ping-pong buffers. 8 waves / block (8 samples).
// ---------------------------------------------------------------------------

template<int CIN, int COUT>
__device__ inline void conv_relu(const float* __restrict__ src,
                                 float* __restrict__ dst, int L,
                                 const float* __restrict__ cw,
                                 const float* __restrict__ cb,
                                 int g, int lane) {
    for (int co = 0; co < COUT; ++co) {
        float w[CIN][5];
#pragma unroll
        for (int ci = 0; ci < CIN; ++ci)
#pragma unroll
            for (int k = 0; k < 5; ++k)
                w[ci][k] = cw[((size_t)(g * COUT + co) * CIN + ci) * 5 + k];
        float bv = cb[g * COUT + co];
        for (int t = lane; t < L; t += 32) {
            float acc = bv;
#pragma unroll
            for (int ci = 0; ci < CIN; ++ci) {
#pragma unroll
                for (int k = 0; k < 5; ++k) {
                    int tt = t + k - 2;
                    float xv = (tt >= 0 && tt < L) ? src[ci * L + tt] : 0.f;
                    acc = fmaf(w[ci][k], xv, acc);
                }
            }
            dst[co * L + t] = fmaxf(acc, 0.f);
        }
    }
}

__device__ inline void maxpool5s2(const float* __restrict__ src,
                                  float* __restrict__ dst,
                                  int L, int Lo, int C, int lane) {
    for (int idx = lane; idx < C * Lo; idx += 32) {
        int c = idx / Lo, j = idx - c * Lo;
        const float* s = src + c * L + 2 * j;
        float m = s[0];
        m = fmaxf(m, s[1]);
        m = fmaxf(m, s[2]);
        m = fmaxf(m, s[3]);
        m = fmaxf(m, s[4]);
        dst[c * Lo + j] = m;
    }
}

// per-wave LDS: buf0 (1186 floats) + buf1 (820 floats) = 2006 floats
#define CONV_WAVE_LDS 2006

__global__ __launch_bounds__(256)
void conv_stack(const float* __restrict__ meteo,   // (B,237,10)
                const float* __restrict__ x_train, // (B,95)
                const float* __restrict__ cw1, const float* __restrict__ cb1,
                const float* __restrict__ cw2, const float* __restrict__ cb2,
                const float* __restrict__ cw3, const float* __restrict__ cb3,
                const float* __restrict__ cw4, const float* __restrict__ cb4,
                const float* __restrict__ cw5, const float* __restrict__ cb5,
                const float* __restrict__ cw6, const float* __restrict__ cb6,
                float* __restrict__ H0)            // (B,3072)
{
    __shared__ float smem[CONV_WAVE_LDS * 8];
    int lane = threadIdx.x & 31;
    int wave = threadIdx.x >> 5;
    int b = blockIdx.x * 8 + wave;
    int g = blockIdx.y;
    float* buf0 = smem + wave * CONV_WAVE_LDS;
    float* buf1 = buf0 + 1186;

    for (int t = lane; t < 237; t += 32)
        buf0[t] = meteo[(size_t)b * 2370 + t * 10 + g];
    __syncthreads();
    conv_relu<1, 3>(buf0, buf1, 237, cw1, cb1, g, lane);  __syncthreads(); // a1 3x237
    conv_relu<3, 5>(buf1, buf0, 237, cw2, cb2, g, lane);  __syncthreads(); // a2 5x237
    maxpool5s2(buf0, buf1, 237, 117, 5, lane);            __syncthreads(); // p1 5x117
    conv_relu<5, 7>(buf1, buf0, 117, cw3, cb3, g, lane);  __syncthreads(); // a3 7x117
    conv_relu<7, 7>(buf0, buf1, 117, cw4, cb4, g, lane);  __syncthreads(); // a4 7x117
    maxpool5s2(buf1, buf0, 117, 57, 7, lane);             __syncthreads(); // p2 7x57
    conv_relu<7, 9>(buf0, buf1, 57, cw5, cb5, g, lane);   __syncthreads(); // a5 9x57
    conv_relu<9, 11>(buf1, buf0, 57, cw6, cb6, g, lane);  __syncthreads(); // a6 11x57
    maxpool5s2(buf0, buf1, 57, 27, 11, lane);             __syncthreads(); // p3 11x27

    float* H0row = H0 + (size_t)b * 3072;
    for (int idx = lane; idx < 297; idx += 32)
        H0row[idx * 10 + g] = buf1[idx];       // feat col = idx*10 + g
    if (g == 0) {
        for (int i = lane; i < 95; i += 32)
            H0row[2970 + i] = x_train[(size_t)b * 95 + i];
        if (lane < 7) H0row[3065 + lane] = 0.f;   // zero pad 3065..3071
    }
}

// ---------------------------------------------------------------------------
// WMMA fp32 GEMM: Out(M x ldOut) = relu(A(M x ldA) * W(Kreal x Nreal) + bias)
// Block tile 128x128, 8 waves (4x2), wave tile 32x64 = 2x4 WMMA 16x16 frags.
// BK=16 (4 WMMA K-steps). LDS double-buffered with register prefetch; one
// barrier per K-chunk. Bs uses a paired layout: element (k,n) lives at
// [(k>>1)*2N + 2n + (k&1)] so each B fragment is one aligned b64 LDS load.
// A is fully padded (ldA mult of 16, pad cols zero) -> unguarded A loads.
// ---------------------------------------------------------------------------

__global__ __launch_bounds__(256)
void gemm_bias_relu(const float* __restrict__ A, int ldA,
                    const float* __restrict__ W,
                    const float* __restrict__ bias,
                    float* __restrict__ Out, int ldOut,
                    int Kreal, int Nreal)
{
    constexpr int BM = 128, BN = 128, BK = 16;
    __shared__ float As[2][BM * BK];          // [m][k], stride BK        (8KB x2)
    __shared__ float Bs[2][(BK / 2) * BN * 2]; // paired [k>>1][n][k&1]   (8KB x2)

    int tid  = threadIdx.x;
    int lane = tid & 31;
    int wave = tid >> 5;
    int wm = wave & 3;      // 4 wave rows  -> m offset wm*32
    int wn = wave >> 2;     // 2 wave cols  -> n offset wn*64
    int l15 = lane & 15;
    int lh  = lane >> 4;    // lane half selects K pair

    int bm0 = blockIdx.x * BM;
    int bn0 = blockIdx.y * BN;

    // staging thread mapping
    int am  = tid >> 2;            // A: rows am, am+64
    int ajc = (tid & 3) * 4;       // A: k sub-offset 0,4,8,12
    int btr = tid >> 5;            // B: k pair row 0..7  (k = 2*btr, 2*btr+1)
    int bn4 = (tid & 31) * 4;      // B: 4 consecutive n

    v8f acc[2][4];
#pragma unroll
    for (int i = 0; i < 2; ++i)
#pragma unroll
        for (int j = 0; j < 4; ++j) acc[i][j] = (v8f){};

    const int KT = ldA / BK;

    // ---- stage tile 0 ----
    {
        *(float4*)&As[0][am * BK + ajc] =
            *(const float4*)(A + (size_t)(bm0 + am) * ldA + ajc);
        *(float4*)&As[0][(am + 64) * BK + ajc] =
            *(const float4*)(A + (size_t)(bm0 + am + 64) * ldA + ajc);
#pragma unroll
        for (int r = 0; r < 2; ++r)
#pragma unroll
            for (int j = 0; j < 4; ++j) {
                int kk = 2 * btr + r, nn = bn4 + j;
                float v = 0.f;
                if (kk < Kreal && (bn0 + nn) < Nreal)
                    v = W[(size_t)kk * Nreal + (bn0 + nn)];
                Bs[0][btr * (2 * BN) + nn * 2 + r] = v;
            }
    }
    __syncthreads();

    for (int kt = 0; kt < KT; ++kt) {
        int cur = kt & 1;
        int nxt = cur ^ 1;
        bool more = (kt + 1) < KT;

        // ---- prefetch next tile into registers ----
        float4 pa0, pa1;
        float pb[2][4];
        if (more) {
            int k0 = (kt + 1) * BK;
            pa0 = *(const float4*)(A + (size_t)(bm0 + am) * ldA + k0 + ajc);
            pa1 = *(const float4*)(A + (size_t)(bm0 + am + 64) * ldA + k0 + ajc);
#pragma unroll
            for (int r = 0; r < 2; ++r)
#pragma unroll
                for (int j = 0; j < 4; ++j) {
                    int kk = k0 + 2 * btr + r, nn = bn0 + bn4 + j;
                    float v = 0.f;
                    if (kk < Kreal && nn < Nreal)
                        v = W[(size_t)kk * Nreal + nn];
                    pb[r][j] = v;
                }
        }

        // ---- compute on current buffer: 4 K-steps x (2x4) WMMA ----
#pragma unroll
        for (int ks = 0; ks < 4; ++ks) {
            int kk = ks * 4 + lh * 2;      // this lane's K pair base (even)
            v2f a0 = *(const v2f*)&As[cur][(wm * 32 + l15) * BK + kk];
            v2f a1 = *(const v2f*)&As[cur][(wm * 32 + 16 + l15) * BK + kk];
            v2f b[4];
#pragma unroll
            for (int s = 0; s < 4; ++s) {
                int col = wn * 64 + s * 16 + l15;
                b[s] = *(const v2f*)&Bs[cur][(ks * 2 + lh) * (2 * BN) + col * 2];
            }
#pragma unroll
            for (int s = 0; s < 4; ++s) {
                acc[0][s] = __builtin_amdgcn_wmma_f32_16x16x4_f32(
                    false, a0, false, b[s], (short)0, acc[0][s], false, false);
                acc[1][s] = __builtin_amdgcn_wmma_f32_16x16x4_f32(
                    false, a1, false, b[s], (short)0, acc[1][s], false, false);
            }
        }

        // ---- commit prefetch to the other buffer ----
        if (more) {
            *(float4*)&As[nxt][am * BK + ajc] = pa0;
            *(float4*)&As[nxt][(am + 64) * BK + ajc] = pa1;
#pragma unroll
            for (int r = 0; r < 2; ++r)
#pragma unroll
                for (int j = 0; j < 4; ++j)
                    Bs[nxt][btr * (2 * BN) + (bn4 + j) * 2 + r] = pb[r][j];
        }
        __syncthreads();
    }

    // ---- epilogue: bias + relu, zero the pad columns ----
#pragma unroll
    for (int s = 0; s < 4; ++s) {
        int n = bn0 + wn * 64 + s * 16 + l15;
        float bia = (n < Nreal) ? bias[n] : 0.f;
        int mbase = bm0 + wm * 32 + lh * 8;
#pragma unroll
        for (int j = 0; j < 8; ++j) {
            int m0 = mbase + j;
            Out[(size_t)m0 * ldOut + n] =
                (n < Nreal) ? fmaxf(acc[0][s][j] + bia, 0.f) : 0.f;
            Out[(size_t)(m0 + 16) * ldOut + n] =
                (n < Nreal) ? fmaxf(acc[1][s][j] + bia, 0.f) : 0.f;
        }
    }
}

// ---------------------------------------------------------------------------
// Final head: out[b] = [h4(20) | x_gboost] . Wc(21) + bc
// ---------------------------------------------------------------------------
__global__ __launch_bounds__(256)
void final_head(const float* __restrict__ h4,   // (B,128) padded
                const float* __restrict__ xg,   // (B,1)
                const float* __restrict__ Wc,   // (21)
                const float* __restrict__ bc,   // (1)
                float* __restrict__ out)        // (B,1)
{
    int b = blockIdx.x * 256 + threadIdx.x;
    float acc = bc[0];
#pragma unroll
    for (int i = 0; i < 20; ++i)
        acc = fmaf(h4[(size_t)b * 128 + i], Wc[i], acc);
    acc = fmaf(xg[b], Wc[20], acc);
    out[b] = acc;
}

// ---------------------------------------------------------------------------

extern "C" void kernel_launch(void* const* d_in, const int* in_sizes, int n_in,
                              void* d_out, int out_size, void* d_ws, size_t ws_size,
                              hipStream_t stream) {
    const float* x_train = (const float*)d_in[0];
    const float* meteo   = (const float*)d_in[1];
    const float* xg      = (const float*)d_in[2];
    const float* cw1 = (const float*)d_in[3];  const float* cb1 = (const float*)d_in[4];
    const float* cw2 = (const float*)d_in[5];  const float* cb2 = (const float*)d_in[6];
    const float* cw3 = (const float*)d_in[7];  const float* cb3 = (const float*)d_in[8];
    const float* cw4 = (const float*)d_in[9];  const float* cb4 = (const float*)d_in[10];
    const float* cw5 = (const float*)d_in[11]; const float* cb5 = (const float*)d_in[12];
    const float* cw6 = (const float*)d_in[13]; const float* cb6 = (const float*)d_in[14];
    const float* W1 = (const float*)d_in[15];  const float* b1 = (const float*)d_in[16];
    const float* W2 = (const float*)d_in[17];  const float* b2 = (const float*)d_in[18];
    const float* W3 = (const float*)d_in[19];  const float* b3 = (const float*)d_in[20];
    const float* W4 = (const float*)d_in[21];  const float* b4 = (const float*)d_in[22];
    const float* Wc = (const float*)d_in[23];  const float* bc = (const float*)d_in[24];
    float* out = (float*)d_out;

    const int B = 4096;
    char* ws = (char*)d_ws;
    // region A (offset 0): H0 (B x 3072) -> later reused for h2 (B x 896), h4 (B x 128)
    // region B (offset OFF1): h1 (B x 1536) -> later reused for h3 (B x 128)
    float* H0 = (float*)(ws);
    size_t OFF1 = (size_t)B * 3072 * sizeof(float);
    float* h1 = (float*)(ws + OFF1);
    float* h2 = (float*)(ws);            // H0 dead after layer 1
    float* h3 = (float*)(ws + OFF1);     // h1 dead after layer 2
    float* h4 = (float*)(ws);            // h2 dead after layer 3

    // 1) conv stacks -> H0 (feat | x_train | zero pad)
    conv_stack<<<dim3(B / 8, 10), 256, 0, stream>>>(
        meteo, x_train, cw1, cb1, cw2, cb2, cw3, cb3,
        cw4, cb4, cw5, cb5, cw6, cb6, H0);

    // 2) MLP layers (WMMA GEMM), padded leading dims 3072 -> 1536 -> 896 -> 128 -> 128
    gemm_bias_relu<<<dim3(B / 128, 1536 / 128), 256, 0, stream>>>(
        H0, 3072, W1, b1, h1, 1536, 3065, 1500);
    gemm_bias_relu<<<dim3(B / 128, 896 / 128), 256, 0, stream>>>(
        h1, 1536, W2, b2, h2, 896, 1500, 800);
    gemm_bias_relu<<<dim3(B / 128, 1), 256, 0, stream>>>(
        h2, 896, W3, b3, h3, 128, 800, 100);
    gemm_bias_relu<<<dim3(B / 128, 1), 256, 0, stream>>>(
        h3, 128, W4, b4, h4, 128, 100, 20);

    // 3) final 21-wide head
    final_head<<<dim3(B / 256), 256, 0, stream>>>(h4, xg, Wc, bc, out);
}